// HNO_6270652252896
// MI455X (gfx1250) — compile-verified
//
#include <hip/hip_runtime.h>
#include <math.h>

#define NN 50000
#define EE 800000
#define HH 128
#define NH (NN * HH)

typedef float v2f __attribute__((ext_vector_type(2)));
typedef float v8f __attribute__((ext_vector_type(8)));

// ---------------------------------------------------------------- utilities
__global__ void init_combine_kernel(float* __restrict__ out,
                                    const float* __restrict__ base,
                                    float beta, int n) {
    int i = blockIdx.x * blockDim.x + threadIdx.x;
    if (i < n) out[i] = base ? beta * base[i] : 0.0f;
}

// Repack W[4][128][128] -> Wp[t][k/2][n][2] so the WMMA B-fragment
// {W[k][n], W[k+1][n]} is a contiguous, 8B-aligned float2.
__global__ void pack_w_kernel(const float* __restrict__ W, float* __restrict__ Wp) {
    int i = blockIdx.x * blockDim.x + threadIdx.x;
    if (i >= 4 * HH * HH) return;
    int t = i >> 14;             // /16384
    int rem = i & 16383;
    int kb = rem >> 7;           // 0..127
    int n = rem & 127;
    int kk2 = kb >> 1;
    int j = kb & 1;
    Wp[(((t << 6) + kk2) * HH + n) * 2 + j] = W[i];
}

// ---------------------------------------------------------------- degree / norm
__global__ void deg_kernel(const int* __restrict__ src, float* __restrict__ deg,
                           int ne) {
    int i = blockIdx.x * blockDim.x + threadIdx.x;
    if (i < ne) atomicAdd(&deg[src[i]], 1.0f);
}

__global__ void dinv_kernel(const float* __restrict__ deg, float* __restrict__ dinv,
                            int n) {
    int i = blockIdx.x * blockDim.x + threadIdx.x;
    if (i < n) {
        float d = deg[i];
        dinv[i] = (d > 0.0f) ? rsqrtf(fmaxf(d, 1.0f)) : 0.0f;
    }
}

__global__ void norm_kernel(const int* __restrict__ src, const int* __restrict__ dst,
                            const float* __restrict__ dinv, float* __restrict__ nw,
                            int ne) {
    int i = blockIdx.x * blockDim.x + threadIdx.x;
    if (i < ne) nw[i] = -dinv[src[i]] * dinv[dst[i]];
}

// ---------------------------------------------------------------- sparse props
// out[dst] += scale * norm * h[src]   (128-wide rows, one wave per edge)
__global__ void prop128_kernel(const float* __restrict__ hin, float* __restrict__ outp,
                               const int* __restrict__ src, const int* __restrict__ dst,
                               const float* __restrict__ nw, float scale, int ne) {
    int tid = blockIdx.x * blockDim.x + threadIdx.x;
    int e = tid >> 5;
    if (e >= ne) return;
    int f = (tid & 31) << 2;
    int s = src[e], d = dst[e];
    float w = scale * nw[e];
    const float4 v = *(const float4*)(hin + s * HH + f);
    float* o = outp + d * HH + f;
    atomicAdd(o + 0, w * v.x);
    atomicAdd(o + 1, w * v.y);
    atomicAdd(o + 2, w * v.z);
    atomicAdd(o + 3, w * v.w);
}

// 3-wide version for layer 1 inputs
__global__ void prop3_kernel(const float* __restrict__ hin, float* __restrict__ outp,
                             const int* __restrict__ src, const int* __restrict__ dst,
                             const float* __restrict__ nw, float scale, int ne) {
    int e = blockIdx.x * blockDim.x + threadIdx.x;
    if (e >= ne) return;
    int s = src[e], d = dst[e];
    float w = scale * nw[e];
    atomicAdd(&outp[d * 3 + 0], w * hin[s * 3 + 0]);
    atomicAdd(&outp[d * 3 + 1], w * hin[s * 3 + 1]);
    atomicAdd(&outp[d * 3 + 2], w * hin[s * 3 + 2]);
}

// ---------------------------------------------------------------- WMMA GEMM
// out[N,128] = sum_t Tt[N,128] @ W[t,128,128] + bias ; fused activation.
// Block = 128 threads (4 waves). Tile: 16 rows x 128 cols, wave w owns cols
// [32w, 32w+32) as two 16x16 accumulators. A tiles staged in LDS (pad 132
// floats/row => conflict-free v2f reads for the 16x4 f32 A-fragment layout).
// B fragments come from the packed layout (single b64 load per fragment).
__global__ __launch_bounds__(128)
void cheb_gemm_wmma_kernel(const float* __restrict__ T0, const float* __restrict__ T1,
                           const float* __restrict__ T2, const float* __restrict__ T3,
                           const float* __restrict__ Wp, const float* __restrict__ bias,
                           float* __restrict__ out, int act) {
    __shared__ float As[16][132];
    const float* Ts[4] = {T0, T1, T2, T3};

    const int tid  = threadIdx.x;
    const int lane = tid & 31;
    const int wave = tid >> 5;
    const int row0 = blockIdx.x * 16;
    const int m    = lane & 15;          // output/input row within tile
    const int kh   = (lane >> 4) << 1;   // 0 (lanes 0-15) or 2 (lanes 16-31)
    const int khp  = lane >> 4;          // k-pair offset for packed B
    const int n0   = (wave << 5) + (lane & 15);
    const int n1   = n0 + 16;

    v8f c0 = {};
    v8f c1 = {};

    for (int t = 0; t < 4; ++t) {
        const float* Tt = Ts[t];
        // stage 16x128 A tile: each thread copies 16 contiguous floats
        {
            int r = tid >> 3;           // 0..15
            int c = (tid & 7) << 4;     // 0,16,...,112
            const float4* sp = (const float4*)(Tt + (row0 + r) * HH + c);
            float4* dp = (float4*)(&As[r][c]);
            dp[0] = sp[0]; dp[1] = sp[1]; dp[2] = sp[2]; dp[3] = sp[3];
        }
        __syncthreads();

        const float* Wt = Wp + t * (64 * HH * 2);
        #pragma unroll 4
        for (int kk = 0; kk < HH; kk += 4) {
            v2f a = *(const v2f*)&As[m][kk + kh];
            const float* wrow = Wt + ((((kk >> 1) + khp) * HH) << 1);
            v2f b0 = *(const v2f*)(wrow + (n0 << 1));
            v2f b1 = *(const v2f*)(wrow + (n1 << 1));
            c0 = __builtin_amdgcn_wmma_f32_16x16x4_f32(false, a, false, b0,
                                                       (short)0, c0, false, false);
            c1 = __builtin_amdgcn_wmma_f32_16x16x4_f32(false, a, false, b1,
                                                       (short)0, c1, false, false);
        }
        __syncthreads();
    }

    // epilogue: C/D layout -> VGPR v holds row (v + 8*(lane>=16)), col = lane%16
    float bias0 = bias[n0];
    float bias1 = bias[n1];
    int rbase = row0 + ((lane >> 4) << 3);
    #pragma unroll
    for (int v = 0; v < 8; ++v) {
        int row = rbase + v;
        float x0 = c0[v] + bias0;
        float x1 = c1[v] + bias1;
        if (act == 1) {            // leaky_relu(0.01)
            x0 = x0 > 0.0f ? x0 : 0.01f * x0;
            x1 = x1 > 0.0f ? x1 : 0.01f * x1;
        } else if (act == 2) {     // relu
            x0 = fmaxf(x0, 0.0f);
            x1 = fmaxf(x1, 0.0f);
        }
        out[row * HH + n0] = x0;
        out[row * HH + n1] = x1;
    }
}

// layer-1 GEMM (in-dim 3): scalar, trivial FLOPs
__global__ void cheb_gemm3_kernel(const float* __restrict__ T0, const float* __restrict__ T1,
                                  const float* __restrict__ T2, const float* __restrict__ T3,
                                  const float* __restrict__ W, const float* __restrict__ bias,
                                  float* __restrict__ out, int act) {
    int i = blockIdx.x * blockDim.x + threadIdx.x;
    if (i >= NH) return;
    int n = i >> 7, f = i & 127;
    const float* Ts[4] = {T0, T1, T2, T3};
    float acc = bias[f];
    #pragma unroll
    for (int t = 0; t < 4; ++t) {
        const float* h = Ts[t] + n * 3;
        const float* w = W + t * 3 * HH + f;
        acc += h[0] * w[0] + h[1] * w[HH] + h[2] * w[2 * HH];
    }
    if (act == 1) acc = acc > 0.0f ? acc : 0.01f * acc;
    else if (act == 2) acc = fmaxf(acc, 0.0f);
    out[i] = acc;
}

// ---------------------------------------------------------------- batch norm
__global__ __launch_bounds__(256)
void bn_stats_kernel(const float* __restrict__ h, float* __restrict__ mean,
                     float* __restrict__ var) {
    __shared__ float ss[256];
    __shared__ float sq[256];
    int f = blockIdx.x;
    float s = 0.0f, q = 0.0f;
    for (int n = threadIdx.x; n < NN; n += 256) {
        float x = h[n * HH + f];
        s += x;
        q += x * x;
    }
    ss[threadIdx.x] = s;
    sq[threadIdx.x] = q;
    __syncthreads();
    for (int st = 128; st > 0; st >>= 1) {
        if (threadIdx.x < st) {
            ss[threadIdx.x] += ss[threadIdx.x + st];
            sq[threadIdx.x] += sq[threadIdx.x + st];
        }
        __syncthreads();
    }
    if (threadIdx.x == 0) {
        float mm = ss[0] / (float)NN;
        mean[f] = mm;
        var[f] = sq[0] / (float)NN - mm * mm;   // biased var
    }
}

__global__ void bn_apply_kernel(float* __restrict__ h, const float* __restrict__ mean,
                                const float* __restrict__ var, const float* __restrict__ g,
                                const float* __restrict__ be, int n) {
    int i = blockIdx.x * blockDim.x + threadIdx.x;
    if (i >= n) return;
    int f = i & 127;
    h[i] = g[f] * (h[i] - mean[f]) * rsqrtf(var[f] + 1e-5f) + be[f];
}

// ---------------------------------------------------------------- readout
// wave-per-node: L2 normalize row, then [1x128] @ [128x3] + br
__global__ __launch_bounds__(256)
void final_kernel(const float* __restrict__ h, const float* __restrict__ Wr,
                  const float* __restrict__ br, float* __restrict__ out) {
    int wid = (blockIdx.x * blockDim.x + threadIdx.x) >> 5;
    int lane = threadIdx.x & 31;
    if (wid >= NN) return;
    int f = lane << 2;
    float4 v = *(const float4*)(h + wid * HH + f);
    float ss = v.x * v.x + v.y * v.y + v.z * v.z + v.w * v.w;
    #pragma unroll
    for (int msk = 16; msk > 0; msk >>= 1) ss += __shfl_xor(ss, msk, 32);
    float inv = 1.0f / fmaxf(sqrtf(ss), 1e-12f);
    float p[3];
    #pragma unroll
    for (int o = 0; o < 3; ++o) {
        p[o] = v.x * Wr[(f + 0) * 3 + o] + v.y * Wr[(f + 1) * 3 + o] +
               v.z * Wr[(f + 2) * 3 + o] + v.w * Wr[(f + 3) * 3 + o];
        #pragma unroll
        for (int msk = 16; msk > 0; msk >>= 1) p[o] += __shfl_xor(p[o], msk, 32);
    }
    if (lane == 0) {
        out[wid * 3 + 0] = inv * p[0] + br[0];
        out[wid * 3 + 1] = inv * p[1] + br[1];
        out[wid * 3 + 2] = inv * p[2] + br[2];
    }
}

// ---------------------------------------------------------------- host side
extern "C" void kernel_launch(void* const* d_in, const int* in_sizes, int n_in,
                              void* d_out, int out_size, void* d_ws, size_t ws_size,
                              hipStream_t stream) {
    (void)in_sizes; (void)n_in; (void)out_size; (void)ws_size;

    const float* x   = (const float*)d_in[0];
    const int* ei    = (const int*)d_in[1];
    const int* src   = ei;
    const int* dst   = ei + EE;
    const float* W1  = (const float*)d_in[2];
    const float* b1  = (const float*)d_in[3];
    const float* W2  = (const float*)d_in[4];
    const float* b2  = (const float*)d_in[5];
    const float* W3  = (const float*)d_in[6];
    const float* b3  = (const float*)d_in[7];
    const float* W4  = (const float*)d_in[8];
    const float* b4  = (const float*)d_in[9];
    const float* g1  = (const float*)d_in[10];
    const float* be1 = (const float*)d_in[11];
    const float* g2  = (const float*)d_in[12];
    const float* be2 = (const float*)d_in[13];
    const float* g3  = (const float*)d_in[14];
    const float* be3 = (const float*)d_in[15];
    const float* Wr  = (const float*)d_in[16];
    const float* br  = (const float*)d_in[17];
    float* out = (float*)d_out;

    float* ws   = (float*)d_ws;
    float* deg  = ws;                 // N
    float* dinv = deg + NN;           // N
    float* nw   = dinv + NN;          // E
    float* bufA = nw + EE;            // N*H (16B-aligned: offset 900000 floats)
    float* bufB = bufA + NH;          // N*H
    float* T1   = bufB + NH;          // N*H
    float* T2   = T1 + NH;            // N*H
    float* T3   = T2 + NH;            // N*H
    float* mean = T3 + NH;            // 128
    float* var  = mean + 128;         // 128
    float* Wpck = var + 128;          // 4*128*128 packed weights

    const int TB = 256;
    auto blk = [](int n, int t) { return (n + t - 1) / t; };

    // ---- degrees and edge weights
    init_combine_kernel<<<blk(NN, TB), TB, 0, stream>>>(deg, nullptr, 0.0f, NN);
    deg_kernel<<<blk(EE, TB), TB, 0, stream>>>(src, deg, EE);
    dinv_kernel<<<blk(NN, TB), TB, 0, stream>>>(deg, dinv, NN);
    norm_kernel<<<blk(EE, TB), TB, 0, stream>>>(src, dst, dinv, nw, EE);

    // ---- layer 1 (in-dim 3), reuse T1/T2/T3 prefixes as N*3 buffers
    const int N3 = NN * 3;
    init_combine_kernel<<<blk(N3, TB), TB, 0, stream>>>(T1, nullptr, 0.0f, N3);
    prop3_kernel<<<blk(EE, TB), TB, 0, stream>>>(x, T1, src, dst, nw, 1.0f, EE);
    init_combine_kernel<<<blk(N3, TB), TB, 0, stream>>>(T2, x, -1.0f, N3);
    prop3_kernel<<<blk(EE, TB), TB, 0, stream>>>(T1, T2, src, dst, nw, 2.0f, EE);
    init_combine_kernel<<<blk(N3, TB), TB, 0, stream>>>(T3, T1, -1.0f, N3);
    prop3_kernel<<<blk(EE, TB), TB, 0, stream>>>(T2, T3, src, dst, nw, 2.0f, EE);
    cheb_gemm3_kernel<<<blk(NH, TB), TB, 0, stream>>>(x, T1, T2, T3, W1, b1, bufA, 1);
    bn_stats_kernel<<<HH, 256, 0, stream>>>(bufA, mean, var);
    bn_apply_kernel<<<blk(NH, TB), TB, 0, stream>>>(bufA, mean, var, g1, be1, NH);

    // ---- layers 2..4 helper
    auto run_cheb128 = [&](const float* hin, const float* W, const float* b,
                           float* outp, int act) {
        pack_w_kernel<<<blk(4 * HH * HH, TB), TB, 0, stream>>>(W, Wpck);
        init_combine_kernel<<<blk(NH, TB), TB, 0, stream>>>(T1, nullptr, 0.0f, NH);
        prop128_kernel<<<blk(EE * 32, TB), TB, 0, stream>>>(hin, T1, src, dst, nw, 1.0f, EE);
        init_combine_kernel<<<blk(NH, TB), TB, 0, stream>>>(T2, hin, -1.0f, NH);
        prop128_kernel<<<blk(EE * 32, TB), TB, 0, stream>>>(T1, T2, src, dst, nw, 2.0f, EE);
        init_combine_kernel<<<blk(NH, TB), TB, 0, stream>>>(T3, T1, -1.0f, NH);
        prop128_kernel<<<blk(EE * 32, TB), TB, 0, stream>>>(T2, T3, src, dst, nw, 2.0f, EE);
        cheb_gemm_wmma_kernel<<<NN / 16, 128, 0, stream>>>(hin, T1, T2, T3, Wpck, b, outp, act);
    };

    // layer 2: leaky relu + bn
    run_cheb128(bufA, W2, b2, bufB, 1);
    bn_stats_kernel<<<HH, 256, 0, stream>>>(bufB, mean, var);
    bn_apply_kernel<<<blk(NH, TB), TB, 0, stream>>>(bufB, mean, var, g2, be2, NH);

    // layer 3: relu + bn
    run_cheb128(bufB, W3, b3, bufA, 2);
    bn_stats_kernel<<<HH, 256, 0, stream>>>(bufA, mean, var);
    bn_apply_kernel<<<blk(NH, TB), TB, 0, stream>>>(bufA, mean, var, g3, be3, NH);

    // layer 4: no act, no bn
    run_cheb128(bufA, W4, b4, bufB, 0);

    // readout: L2-normalize + 128->3 projection
    final_kernel<<<blk(NN * 32, TB), TB, 0, stream>>>(bufB, Wr, br, out);
}